// SpectralSPE_67894843015186
// MI455X (gfx1250) — compile-verified
//
#include <hip/hip_runtime.h>
#include <hip/hip_bf16.h>

// ---------------------------------------------------------------------------
// SpectralSPE on MI455X (gfx1250).
//
// Key identity: istft(relu(psd) * stft(x)) per frame is
//   y_frame = w ⊙ (h_d ⊛ (w ⊙ x_frame)),   h_d = irfft(relu(psd_d), 400)
// (the stft's N^-0.5 and istft's N^+0.5 cancel), i.e. a dense 400x400 matrix
//   M_d[j,i] = w[j] * h_d[(i-j) mod 400] * w[i]
// applied per frame, then overlap-add and divide by sum-of-squared-windows.
// This maps the whole filter to f32 WMMA GEMMs: Y[41,400] = Frames[41,400]*M_d.
// Columns padded 400->416 so each wave runs TWO independent WMMA accumulator
// chains off one shared A fragment.
// ---------------------------------------------------------------------------

typedef float v2f __attribute__((ext_vector_type(2)));
typedef float v8f __attribute__((ext_vector_type(8)));

#define D_CH   64
#define NFFT   400
#define NFFTP  416   // padded column count (26 col tiles -> 13 pairs)
#define NFREQ  201
#define HOP    50
#define NB     4
#define NUMM   16
#define LEN    2048
#define PAD    200
#define XPLEN  2448   // LEN + 2*PAD
#define TOTAL  2400   // NFFT + (TFRAMES-1)*HOP
#define YPADLEN 2432  // TOTAL + room for zero pad-column OLA (50*40+415 = 2415)
#define TFRAMES 41
#define SIGS   (NB * NUMM)   // 64 signals per channel d

// Workspace layout (float units). Total ~27.5M floats (~110 MB).
#define OFF_MP     0                           // float2[64][200][416] (pair-interleaved M_d)
#define MP_FLOATS  (D_CH * 200 * NFFTP * 2)
#define OFF_H      (OFF_MP + MP_FLOATS)        // h[64][400]
#define H_FLOATS   (D_CH * NFFT)
#define OFF_W      (OFF_H + H_FLOATS)          // w[400]
#define W_FLOATS   NFFT
#define OFF_IWSQ   (OFF_W + W_FLOATS)          // 1/wsq[2400]
#define IWSQ_FLOATS TOTAL
#define OFF_PEQ    (OFF_IWSQ + IWSQ_FLOATS)    // pe_q[64][4][16][2048]
#define PE_FLOATS  (D_CH * NB * NUMM * LEN)
#define OFF_PEK    (OFF_PEQ + PE_FLOATS)       // pe_k[64][4][16][2048]

__device__ __forceinline__ float hammingf(int k) {
    const float TWO_PI = 6.28318530717958647692f;
    return 0.54f - 0.46f * cosf(TWO_PI * (float)k / (float)NFFT);
}

// ---------------------------------------------------------------------------
// Kernel 0: window w[400], h_d = irfft(relu(psd_d)), inv_wsq[2400].
// ---------------------------------------------------------------------------
__global__ __launch_bounds__(256) void spe_setup(const float* __restrict__ psd,
                                                 float* __restrict__ ws) {
    const int blk = blockIdx.x;
    const int tid = threadIdx.x;
    const float TWO_PI = 6.28318530717958647692f;
    if (blk < D_CH) {
        const int d = blk;
        const float p0 = fmaxf(psd[d * NFREQ + 0], 0.f);
        const float pN = fmaxf(psd[d * NFREQ + 200], 0.f);
        for (int t = tid; t < NFFT; t += blockDim.x) {
            float acc = p0 + pN * ((t & 1) ? -1.f : 1.f);
            const float ang0 = TWO_PI * (float)t / (float)NFFT;
            for (int f = 1; f < 200; ++f) {
                float pf = fmaxf(psd[d * NFREQ + f], 0.f);
                acc += 2.f * pf * cosf(ang0 * (float)f);
            }
            ws[OFF_H + d * NFFT + t] = acc * (1.f / (float)NFFT);
        }
    } else if (blk == D_CH) {
        for (int k = tid; k < NFFT; k += blockDim.x)
            ws[OFF_W + k] = hammingf(k);
    } else {
        for (int p = tid; p < TOTAL; p += blockDim.x) {
            float s = 0.f;
            for (int t = 0; t < TFRAMES; ++t) {
                int r = p - HOP * t;
                if (r >= 0 && r < NFFT) { float wv = hammingf(r); s += wv * wv; }
            }
            ws[OFF_IWSQ + p] = 1.f / s;
        }
    }
}

// ---------------------------------------------------------------------------
// Kernel 1: build pair-interleaved, column-padded M_d:
//   Mp[d][jp][i] = { M_d[2jp][i], M_d[2jp+1][i] },  i in [0,416), zero for i>=400
// ---------------------------------------------------------------------------
__global__ __launch_bounds__(256) void spe_build_m(float* __restrict__ ws) {
    const int d   = blockIdx.y;
    const int idx = blockIdx.x * blockDim.x + threadIdx.x;
    if (idx >= 200 * NFFTP) return;
    const int jp = idx / NFFTP;
    const int i  = idx % NFFTP;
    float2 v = make_float2(0.f, 0.f);
    if (i < NFFT) {
        const float* w = ws + OFF_W;
        const float* h = ws + OFF_H + d * NFFT;
        const int j0 = 2 * jp, j1 = 2 * jp + 1;
        const float wi = w[i];
        v.x = w[j0] * wi * h[(i - j0 + NFFT) % NFFT];
        v.y = w[j1] * wi * h[(i - j1 + NFFT) % NFFT];
    }
    ((float2*)(ws + OFF_MP))[(size_t)(d * 200 + jp) * NFFTP + i] = v;
}

// ---------------------------------------------------------------------------
// Kernel 2: per (d, signal, which-z): frames GEMM via f32 WMMA + OLA in LDS.
//   which==0: filter z_n  -> pe_q (+ 0.25*z_m)
//   which==1: filter z_m  -> pe_k (+ 0.25*z_n)
// 8 waves/WG. Each wave task = 16 frames x 32 cols: two independent WMMA
// accumulator chains sharing one A fragment per K step.
// ---------------------------------------------------------------------------
__global__ __launch_bounds__(256) void spe_filter(const float* __restrict__ z_m,
                                                  const float* __restrict__ z_n,
                                                  float* __restrict__ ws) {
    __shared__ float xp[2800];       // padded signal + zero tail (frame rows 41..47)
    __shared__ float ypad[YPADLEN];  // overlap-add accumulator (+pad-col spill)

    const int bx    = blockIdx.x;
    const int d     = bx >> 7;
    const int rem   = bx & 127;
    const int which = rem >> 6;
    const int sig   = rem & 63;
    const float* zsrc = which ? z_m : z_n;
    const float* zoth = which ? z_n : z_m;
    const size_t sigoff = (size_t)(d * SIGS + sig) * LEN;

    const int tid = threadIdx.x;

    // Reflect-pad (numpy 'reflect', no edge repeat), pre-scale by 1/sqrt(num)=0.25.
    for (int q = tid; q < 2800; q += 256) {
        float v = 0.f;
        if (q < XPLEN) {
            int src = q - PAD;
            if (src < 0) src = -src;
            if (src > LEN - 1) src = 2 * (LEN - 1) - src;
            v = 0.25f * zsrc[sigoff + src];
        }
        xp[q] = v;
    }
    for (int p = tid; p < YPADLEN; p += 256) ypad[p] = 0.f;
    __syncthreads();

    const int wave = tid >> 5;
    const int lane = tid & 31;
    const int hl   = (lane >= 16) ? 1 : 0;  // lane half selects K pair
    const int lm   = lane & 15;
    const float2* Mp = ((const float2*)(ws + OFF_MP)) + (size_t)d * 200 * NFFTP;

    // 3 row-tiles (48 >= 41 frames) x 13 col-pairs = 39 tasks.
    for (int task = wave; task < 39; task += 8) {
        const int rowTile = task / 13;
        const int cp      = task % 13;
        const int col1 = cp * 32 + lm;           // first 16-col tile
        const int col2 = col1 + 16;              // second 16-col tile (may be pad)
        const int tfr  = rowTile * 16 + lm;      // A row (frame index)
        v8f acc1 = {};
        v8f acc2 = {};
        #pragma unroll 4
        for (int kb = 0; kb < NFFT; kb += 4) {
            const int k0 = kb + 2 * hl;
            // A 16x4 f32: lanes 0-15 hold K={kb,kb+1}, lanes 16-31 K={kb+2,kb+3}
            v2f a = *(const v2f*)&xp[HOP * tfr + k0];
            // B 4x16 f32: same K split across lane halves, pair pre-interleaved
            const int jp = (kb >> 1) + hl;
            float2 b1t = Mp[(size_t)jp * NFFTP + col1];
            float2 b2t = Mp[(size_t)jp * NFFTP + col2];
            v2f b1; b1.x = b1t.x; b1.y = b1t.y;
            v2f b2; b2.x = b2t.x; b2.y = b2t.y;
            acc1 = __builtin_amdgcn_wmma_f32_16x16x4_f32(
                false, a, false, b1, (short)0, acc1, false, false);
            acc2 = __builtin_amdgcn_wmma_f32_16x16x4_f32(
                false, a, false, b2, (short)0, acc2, false, false);
        }
        // C/D layout: VGPR r -> frame rowTile*16 + r + 8*hl, col = lm. OLA into LDS.
        // Pad columns (>=400) hold exact zeros; their adds land in ypad[400..2415].
        const int frBase = rowTile * 16 + 8 * hl;
        if (rowTile < 2) {
            // Frames 0..31 are always < 41: no predication, plain ds_add_f32.
            #pragma unroll
            for (int r = 0; r < 8; ++r) {
                atomicAdd(&ypad[HOP * (frBase + r) + col1], acc1[r]);
                atomicAdd(&ypad[HOP * (frBase + r) + col2], acc2[r]);
            }
        } else {
            #pragma unroll
            for (int r = 0; r < 8; ++r) {
                const int fr = frBase + r;
                if (fr < TFRAMES) {
                    atomicAdd(&ypad[HOP * fr + col1], acc1[r]);
                    atomicAdd(&ypad[HOP * fr + col2], acc2[r]);
                }
            }
        }
    }
    __syncthreads();

    // Normalize by sum-of-squared-windows, crop pad, fuse + 0.25*z_other.
    const float* iwsq = ws + OFF_IWSQ;
    float* pe = ws + (which ? OFF_PEK : OFF_PEQ) + sigoff;
    for (int n = tid; n < LEN; n += 256) {
        const int p = n + PAD;
        pe[n] = ypad[p] * iwsq[p] + 0.25f * zoth[sigoff + n];
    }
}

// ---------------------------------------------------------------------------
// Kernel 3: qhat[b,n,m] = sum_d pe[d,b,m,n] * qk[b,n,d]  (34 MFLOP, mem-light)
// ---------------------------------------------------------------------------
__global__ __launch_bounds__(128) void spe_reduce(const float* __restrict__ qk,
                                                  const float* __restrict__ pe,
                                                  float* __restrict__ out) {
    const int b = blockIdx.y;
    const int n = blockIdx.x * 128 + threadIdx.x;
    float acc[NUMM];
    #pragma unroll
    for (int m = 0; m < NUMM; ++m) acc[m] = 0.f;
    for (int d = 0; d < D_CH; ++d) {
        const float s = qk[((size_t)b * LEN + n) * D_CH + d];
        const float* prow = pe + (size_t)(d * NB + b) * NUMM * LEN + n;
        #pragma unroll
        for (int m = 0; m < NUMM; ++m) acc[m] += prow[(size_t)m * LEN] * s;
    }
    float* o = out + ((size_t)b * LEN + n) * NUMM;
    #pragma unroll
    for (int m = 0; m < NUMM; ++m) o[m] = acc[m];
}

// ---------------------------------------------------------------------------
extern "C" void kernel_launch(void* const* d_in, const int* in_sizes, int n_in,
                              void* d_out, int out_size, void* d_ws, size_t ws_size,
                              hipStream_t stream) {
    (void)in_sizes; (void)n_in; (void)out_size; (void)ws_size;
    const float* queries = (const float*)d_in[0];  // [4,2048,64]
    const float* keys    = (const float*)d_in[1];  // [4,2048,64]
    const float* psd     = (const float*)d_in[2];  // [64,201]
    const float* z_m     = (const float*)d_in[3];  // [64,4,16,2048]
    const float* z_n     = (const float*)d_in[4];  // [64,4,16,2048]
    float* ws  = (float*)d_ws;
    float* out = (float*)d_out;

    spe_setup  <<<dim3(D_CH + 2), dim3(256), 0, stream>>>(psd, ws);
    spe_build_m<<<dim3((200 * NFFTP + 255) / 256, D_CH), dim3(256), 0, stream>>>(ws);
    spe_filter <<<dim3(D_CH * 128), dim3(256), 0, stream>>>(z_m, z_n, ws);
    spe_reduce <<<dim3(16, NB), dim3(128), 0, stream>>>(queries, ws + OFF_PEQ, out);
    spe_reduce <<<dim3(16, NB), dim3(128), 0, stream>>>(keys,    ws + OFF_PEK,
                                                        out + NB * LEN * NUMM);
}